// GCN_51951924412872
// MI455X (gfx1250) — compile-verified
//
#include <hip/hip_runtime.h>

typedef float v2f __attribute__((ext_vector_type(2)));
typedef float v8f __attribute__((ext_vector_type(8)));

// ---------------------------------------------------------------------------
// GCN on [N,1] features collapses to scalar gather/scatter per layer:
//   dis = rsqrt(deg+1); p = dis*x
//   s_i = dis_i * (p_i + sum_{e: col=i} p[row_e])          (layer-1 aggregate)
//   t_i = sum_k relu(W1[k]*s_i + b1[k]) * W2[k]            (per-node MLP, WMMA)
//   q   = dis*t
//   out_i = dis_i * (q_i + sum_{e: col=i} q[row_e]) + b2   (layer-2 aggregate)
// ---------------------------------------------------------------------------

__global__ void gcn_zero(float* __restrict__ deg, float* __restrict__ acc1,
                         float* __restrict__ acc2, int n) {
  int i = blockIdx.x * blockDim.x + threadIdx.x;
  if (i < n) { deg[i] = 0.f; acc1[i] = 0.f; acc2[i] = 0.f; }
}

__global__ void gcn_degree(const int* __restrict__ col, float* __restrict__ deg, int E) {
  int i = blockIdx.x * blockDim.x + threadIdx.x;
  int e0 = i * 4;
  if (e0 + 3 < E) {
    int4 c = *reinterpret_cast<const int4*>(col + e0);
    atomicAdd(&deg[c.x], 1.0f);
    atomicAdd(&deg[c.y], 1.0f);
    atomicAdd(&deg[c.z], 1.0f);
    atomicAdd(&deg[c.w], 1.0f);
  } else {
    for (int e = e0; e < E; ++e) atomicAdd(&deg[col[e]], 1.0f);
  }
}

__global__ void gcn_prep(const float* __restrict__ x, const float* __restrict__ deg,
                         float* __restrict__ dis, float* __restrict__ p, int n) {
  int i = blockIdx.x * blockDim.x + threadIdx.x;
  if (i < n) {
    float d = deg[i] + 1.0f;        // self loop
    float r = rsqrtf(d);            // deg >= 1 always
    dis[i] = r;
    p[i] = r * x[i];
  }
}

__global__ void gcn_scatter(const int* __restrict__ row, const int* __restrict__ col,
                            const float* __restrict__ val, float* __restrict__ acc, int E) {
  int i = blockIdx.x * blockDim.x + threadIdx.x;
  int e0 = i * 4;
  if (e0 + 3 < E) {
    int4 r = *reinterpret_cast<const int4*>(row + e0);
    int4 c = *reinterpret_cast<const int4*>(col + e0);
    float v0 = val[r.x], v1 = val[r.y], v2 = val[r.z], v3 = val[r.w];
    atomicAdd(&acc[c.x], v0);
    atomicAdd(&acc[c.y], v1);
    atomicAdd(&acc[c.z], v2);
    atomicAdd(&acc[c.w], v3);
  } else {
    for (int e = e0; e < E; ++e) atomicAdd(&acc[col[e]], val[row[e]]);
  }
}

// Per-node MLP via V_WMMA_F32_16X16X4_F32, 16 nodes per wave.
// Stage 1: H(16x16) = A(16x4:[s,1,0,0]) x B(4x16:[W1;b1;0;0])   -> pre-activation
// Stage 2: T = sum_c A'(16x4: W2 chunk broadcast) x B'(4x16: H^T chunk from LDS)
//          -> every lane l holds t[node l&15].
__global__ void gcn_node_update(const float* __restrict__ dis, const float* __restrict__ p,
                                const float* __restrict__ acc1,
                                const float* __restrict__ W1, const float* __restrict__ b1,
                                const float* __restrict__ W2,
                                float* __restrict__ q, int n) {
  __shared__ float hbuf[8 * 256];                  // 8 waves/block * 16x16 tile
  const int lane = threadIdx.x & 31;
  const int wv = threadIdx.x >> 5;
  const int base = (blockIdx.x * (blockDim.x >> 5) + wv) * 16;
  const int m = lane & 15;
  const bool hi = lane >= 16;
  int node = base + m;
  int cn = node < n ? node : n - 1;                // clamp; masked at store
  float s = dis[cn] * (acc1[cn] + p[cn]);

  // A layout (16x4 f32): lanes0-15 = M, v0:{K0|K2}, v1:{K1|K3} across halves.
  v2f a = { hi ? 0.f : s,      hi ? 0.f : 1.f };
  v2f b = { hi ? 0.f : W1[m],  hi ? 0.f : b1[m] };
  v8f c = {0.f, 0.f, 0.f, 0.f, 0.f, 0.f, 0.f, 0.f};
  v8f h = __builtin_amdgcn_wmma_f32_16x16x4_f32(false, a, false, b, (short)0, c,
                                                false, false);
  // D layout: lane l, vgpr v -> H[node = v + 8*(l>=16)][feat = l&15]
  float* t0 = &hbuf[wv * 256];
#pragma unroll
  for (int i = 0; i < 8; ++i) {
    float hv = h[i] > 0.f ? h[i] : 0.f;            // relu
    t0[(i + (hi ? 8 : 0)) * 16 + m] = hv;          // row-major H[node][feat]
  }
  __syncthreads();

  v8f t = {0.f, 0.f, 0.f, 0.f, 0.f, 0.f, 0.f, 0.f};
#pragma unroll
  for (int ck = 0; ck < 4; ++ck) {
    int k0 = 4 * ck + (hi ? 2 : 0);
    v2f ap = { W2[k0], W2[k0 + 1] };               // A'[m,k] = W2[k] (bcast over M)
    v2f bp = { t0[m * 16 + k0], t0[m * 16 + k0 + 1] };  // B'[k,n] = H[n,k]
    t = __builtin_amdgcn_wmma_f32_16x16x4_f32(false, ap, false, bp, (short)0, t,
                                              false, false);
  }
  // T[m,n] = t[n] for all m; every lane has t[lane&15] in t[0].
  if (!hi && node < n) q[node] = dis[node] * t[0];
}

__global__ void gcn_finalize(const float* __restrict__ dis, const float* __restrict__ q,
                             const float* __restrict__ acc2, const float* __restrict__ b2,
                             float* __restrict__ out, int n) {
  int i = blockIdx.x * blockDim.x + threadIdx.x;
  if (i < n) out[i] = dis[i] * (acc2[i] + q[i]) + b2[0];
}

extern "C" void kernel_launch(void* const* d_in, const int* in_sizes, int n_in,
                              void* d_out, int out_size, void* d_ws, size_t ws_size,
                              hipStream_t stream) {
  const float* x  = (const float*)d_in[0];
  const int*   ei = (const int*)d_in[1];     // edge_index [2, E], int32 per harness ABI
  const float* W1 = (const float*)d_in[2];
  const float* b1 = (const float*)d_in[3];
  const float* W2 = (const float*)d_in[4];
  const float* b2 = (const float*)d_in[5];
  float* out = (float*)d_out;

  const int n = in_sizes[0];                 // N_NODES (x is [N,1])
  const int E = in_sizes[1] / 2;
  const int* row = ei;                       // source j
  const int* col = ei + E;                   // target i

  float* ws   = (float*)d_ws;                // 6*N floats = 2.4 MB
  float* deg  = ws;
  float* dis  = ws + 1 * (size_t)n;
  float* p    = ws + 2 * (size_t)n;
  float* acc1 = ws + 3 * (size_t)n;
  float* q    = ws + 4 * (size_t)n;
  float* acc2 = ws + 5 * (size_t)n;

  const int TB = 256;
  const int nodeBlocks = (n + TB - 1) / TB;
  const int edgeLanes  = (E + 3) / 4;        // 4 edges per lane (int4 loads)
  const int edgeBlocks = (edgeLanes + TB - 1) / TB;
  const int tiles      = (n + 15) / 16;      // 16 nodes per wave
  const int mlpBlocks  = (tiles + 7) / 8;    // 8 waves per block

  gcn_zero       <<<nodeBlocks, TB, 0, stream>>>(deg, acc1, acc2, n);
  gcn_degree     <<<edgeBlocks, TB, 0, stream>>>(col, deg, E);
  gcn_prep       <<<nodeBlocks, TB, 0, stream>>>(x, deg, dis, p, n);
  gcn_scatter    <<<edgeBlocks, TB, 0, stream>>>(row, col, p, acc1, E);
  gcn_node_update<<<mlpBlocks,  TB, 0, stream>>>(dis, p, acc1, W1, b1, W2, q, n);
  gcn_scatter    <<<edgeBlocks, TB, 0, stream>>>(row, col, q, acc2, E);
  gcn_finalize   <<<nodeBlocks, TB, 0, stream>>>(dis, q, acc2, b2, out, n);
}